// Arap_57741540327998
// MI455X (gfx1250) — compile-verified
//
#include <hip/hip_runtime.h>

typedef float v2f __attribute__((ext_vector_type(2)));
typedef float v8f __attribute__((ext_vector_type(8)));

#define ARAP_K 32

// ---------------------------------------------------------------------------
// Wave-wide (32 lane) float sum, returned to all lanes.
// CDNA5 path: V_WMMA_F32_16X16X4_F32 with B == all-ones.
//   D[m][n] = sum_k A[m][k] * 1  ->  every lane's 8 D-VGPRs hold row-sums of
//   half the rows; per-lane add of the 8 regs + shfl_xor(16) = total of all
//   64 A slots.  We place x in A-vgpr0 and 0 in A-vgpr1, so the total is the
//   32-lane sum of x -- independent of the exact A striping (bijection).
// ---------------------------------------------------------------------------
__device__ __forceinline__ float wave_sum32(float x) {
#if defined(__gfx1250__) && __has_builtin(__builtin_amdgcn_wmma_f32_16x16x4_f32)
  v2f a; a[0] = x;    a[1] = 0.0f;
  v2f b; b[0] = 1.0f; b[1] = 1.0f;
  v8f c = {0.f, 0.f, 0.f, 0.f, 0.f, 0.f, 0.f, 0.f};
  c = __builtin_amdgcn_wmma_f32_16x16x4_f32(false, a, false, b, (short)0, c,
                                            false, false);
  float h = ((c[0] + c[1]) + (c[2] + c[3])) + ((c[4] + c[5]) + (c[6] + c[7]));
  h += __shfl_xor(h, 16, 32);
  return h;
#else
  #pragma unroll
  for (int m = 16; m; m >>= 1) x += __shfl_xor(x, m, 32);
  return x;
#endif
}

// One cyclic-Jacobi rotation on symmetric S (pair p,q), accumulating V columns.
#define JROT(SPP, SQQ, SPQ, SXP, SXQ, VPX, VPY, VPZ, VQX, VQY, VQZ)        \
  do {                                                                     \
    float apq = SPQ;                                                       \
    if (fabsf(apq) > 1e-30f) {                                             \
      float tau = (SQQ - SPP) * 0.5f / apq;                                \
      float t = copysignf(1.0f, tau) /                                     \
                (fabsf(tau) + sqrtf(fmaf(tau, tau, 1.0f)));                \
      float ci = rsqrtf(fmaf(t, t, 1.0f));                                 \
      float si = t * ci;                                                   \
      SPP = SPP - t * apq;                                                 \
      SQQ = SQQ + t * apq;                                                 \
      SPQ = 0.0f;                                                          \
      float xp = SXP, xq = SXQ;                                            \
      SXP = ci * xp - si * xq;                                             \
      SXQ = si * xp + ci * xq;                                             \
      float p0, q0;                                                        \
      p0 = VPX; q0 = VQX; VPX = ci * p0 - si * q0; VQX = si * p0 + ci * q0;\
      p0 = VPY; q0 = VQY; VPY = ci * p0 - si * q0; VQY = si * p0 + ci * q0;\
      p0 = VPZ; q0 = VQZ; VPZ = ci * p0 - si * q0; VQZ = si * p0 + ci * q0;\
    }                                                                      \
  } while (0)

#define CSWAP(LA, LB, AX, AY, AZ, BX, BY, BZ)                              \
  do {                                                                     \
    if (LA < LB) {                                                         \
      float tm;                                                            \
      tm = LA; LA = LB; LB = tm;                                           \
      tm = AX; AX = BX; BX = tm;                                           \
      tm = AY; AY = BY; BY = tm;                                           \
      tm = AZ; AZ = BZ; BZ = tm;                                           \
    }                                                                      \
  } while (0)

// One wave per vertex; lane k handles neighbor k (K == 32 == wave32).
__global__ __launch_bounds__(256) void arap_vertex_kernel(
    const float* __restrict__ V, const float* __restrict__ Vd,
    const int* __restrict__ nbrs, const float* __restrict__ wgts,
    float* __restrict__ pv, float* __restrict__ out, int N, int use_atomic) {
  const int lane = threadIdx.x & 31;
  const int wid  = threadIdx.x >> 5;
  const int n = blockIdx.x * (blockDim.x >> 5) + wid;
  if (n >= N) return;  // wave-uniform: EXEC stays all-ones for live waves

  // Center vertex (uniform across the wave; L0 broadcasts).
  const float cx = V[3 * n + 0],  cy = V[3 * n + 1],  cz = V[3 * n + 2];
  const float gx = Vd[3 * n + 0], gy = Vd[3 * n + 1], gz = Vd[3 * n + 2];

  // Per-lane neighbor data (coalesced 128B rows + L2-resident gathers).
  const int   nb = nbrs[n * ARAP_K + lane];
  const float w  = wgts[n * ARAP_K + lane];

  const float ex = V[3 * nb + 0] - cx;   // base edge
  const float ey = V[3 * nb + 1] - cy;
  const float ez = V[3 * nb + 2] - cz;
  const float rx = Vd[3 * nb + 0] - gx;  // deformed edge (rot_hood)
  const float ry = Vd[3 * nb + 1] - gy;
  const float rz = Vd[3 * nb + 2] - gz;
  const float tx = ex * w, ty = ey * w, tz = ez * w;  // tru_hood column

  // tmp[i][j] = sum_k rot_i * tru_j  (9 wave sums via WMMA reducer)
  float a00 = wave_sum32(rx * tx), a01 = wave_sum32(rx * ty), a02 = wave_sum32(rx * tz);
  float a10 = wave_sum32(ry * tx), a11 = wave_sum32(ry * ty), a12 = wave_sum32(ry * tz);
  float a20 = wave_sum32(rz * tx), a21 = wave_sum32(rz * ty), a22 = wave_sum32(rz * tz);

  // ---- 3x3 SVD: eigendecompose S = A^T A with cyclic Jacobi ----
  float s00 = a00 * a00 + a10 * a10 + a20 * a20;
  float s01 = a00 * a01 + a10 * a11 + a20 * a21;
  float s02 = a00 * a02 + a10 * a12 + a20 * a22;
  float s11 = a01 * a01 + a11 * a11 + a21 * a21;
  float s12 = a01 * a02 + a11 * a12 + a21 * a22;
  float s22 = a02 * a02 + a12 * a12 + a22 * a22;

  float v0x = 1.f, v0y = 0.f, v0z = 0.f;   // V columns (right singular vecs)
  float v1x = 0.f, v1y = 1.f, v1z = 0.f;
  float v2x = 0.f, v2y = 0.f, v2z = 1.f;

  #pragma unroll
  for (int sw = 0; sw < 5; ++sw) {
    JROT(s00, s11, s01, s02, s12, v0x, v0y, v0z, v1x, v1y, v1z);
    JROT(s00, s22, s02, s01, s12, v0x, v0y, v0z, v2x, v2y, v2z);
    JROT(s11, s22, s12, s01, s02, v1x, v1y, v1z, v2x, v2y, v2z);
  }

  // Sort eigenvalues descending (LAPACK order), carrying V columns.
  CSWAP(s00, s11, v0x, v0y, v0z, v1x, v1y, v1z);
  CSWAP(s00, s22, v0x, v0y, v0z, v2x, v2y, v2z);
  CSWAP(s11, s22, v1x, v1y, v1z, v2x, v2y, v2z);

  // U columns: u_i = normalize(A v_i); u2 via cross (sign-invariant under det fix).
  float u0x = a00 * v0x + a01 * v0y + a02 * v0z;
  float u0y = a10 * v0x + a11 * v0y + a12 * v0z;
  float u0z = a20 * v0x + a21 * v0y + a22 * v0z;
  float n0 = u0x * u0x + u0y * u0y + u0z * u0z;
  if (n0 > 1e-24f) {
    float iv = rsqrtf(n0); u0x *= iv; u0y *= iv; u0z *= iv;
  } else { u0x = 1.f; u0y = 0.f; u0z = 0.f; }

  float u1x = a00 * v1x + a01 * v1y + a02 * v1z;
  float u1y = a10 * v1x + a11 * v1y + a12 * v1z;
  float u1z = a20 * v1x + a21 * v1y + a22 * v1z;
  float d01 = u0x * u1x + u0y * u1y + u0z * u1z;
  u1x -= d01 * u0x; u1y -= d01 * u0y; u1z -= d01 * u0z;
  float n1 = u1x * u1x + u1y * u1y + u1z * u1z;
  if (n1 > 1e-24f) {
    float iv = rsqrtf(n1); u1x *= iv; u1y *= iv; u1z *= iv;
  } else {
    float axx = (fabsf(u0x) < 0.9f) ? 1.f : 0.f;
    float axy = 1.f - axx;
    u1x = u0y * 0.f - u0z * axy;   // cross(u0, axis)
    u1y = u0z * axx - u0x * 0.f;
    u1z = u0x * axy - u0y * axx;
    float iv = rsqrtf(fmaxf(u1x * u1x + u1y * u1y + u1z * u1z, 1e-24f));
    u1x *= iv; u1y *= iv; u1z *= iv;
  }
  float u2x = u0y * u1z - u0z * u1y;
  float u2y = u0z * u1x - u0x * u1z;
  float u2z = u0x * u1y - u0y * u1x;

  // Reference: R1 = U @ Vh^T = U @ V  (V has right singular vectors as columns)
  float r00 = u0x * v0x + u1x * v0y + u2x * v0z;
  float r01 = u0x * v1x + u1x * v1y + u2x * v1z;
  float r02 = u0x * v2x + u1x * v2y + u2x * v2z;
  float r10 = u0y * v0x + u1y * v0y + u2y * v0z;
  float r11 = u0y * v1x + u1y * v1y + u2y * v1z;
  float r12 = u0y * v2x + u1y * v2y + u2y * v2z;
  float r20 = u0z * v0x + u1z * v0y + u2z * v0z;
  float r21 = u0z * v1x + u1z * v1y + u2z * v1z;
  float r22 = u0z * v2x + u1z * v2y + u2z * v2z;

  float det = r00 * (r11 * r22 - r12 * r21)
            - r01 * (r10 * r22 - r12 * r20)
            + r02 * (r10 * r21 - r11 * r20);

  // Scale last row of Vt by det and recompute: R += (det-1) * u2 (x) V_row2
  const float dm1 = det - 1.0f;
  r00 += dm1 * u2x * v0z;  r01 += dm1 * u2x * v1z;  r02 += dm1 * u2x * v2z;
  r10 += dm1 * u2y * v0z;  r11 += dm1 * u2y * v1z;  r12 += dm1 * u2y * v2z;
  r20 += dm1 * u2z * v0z;  r21 += dm1 * u2z * v1z;  r22 += dm1 * u2z * v2z;

  // arap_k = w * | rot_k - R * edge_k |^2   (w==0 kills padded slots exactly)
  const float dfx = rx - (r00 * ex + r01 * ey + r02 * ez);
  const float dfy = ry - (r10 * ex + r11 * ey + r12 * ez);
  const float dfz = rz - (r20 * ex + r21 * ey + r22 * ez);
  const float arap = w * (dfx * dfx + dfy * dfy + dfz * dfz);

  const float vert = wave_sum32(arap);
  if (lane == 0) {
    if (use_atomic) {
      atomicAdd(out, vert / (float)N);
    } else {
      pv[n] = vert;
    }
  }
}

// Deterministic single-block reduction: fixed traversal order every call.
__global__ __launch_bounds__(256) void arap_reduce_kernel(
    const float* __restrict__ pv, float* __restrict__ out, int N) {
  __shared__ float sm[8];
  float acc = 0.f;
  for (int i = threadIdx.x; i < N; i += 256) acc += pv[i];
  const float ws = wave_sum32(acc);  // all 256 threads active here (EXEC ok)
  const int lane = threadIdx.x & 31, wid = threadIdx.x >> 5;
  if (lane == 0) sm[wid] = ws;
  __syncthreads();
  if (threadIdx.x == 0) {
    float t = 0.f;
    #pragma unroll
    for (int i = 0; i < 8; ++i) t += sm[i];
    out[0] = t / (float)N;
  }
}

extern "C" void kernel_launch(void* const* d_in, const int* in_sizes, int n_in,
                              void* d_out, int out_size, void* d_ws,
                              size_t ws_size, hipStream_t stream) {
  const float* V    = (const float*)d_in[0];
  const float* Vd   = (const float*)d_in[1];
  const int*   nbrs = (const int*)d_in[2];
  const float* wgts = (const float*)d_in[3];
  float* out = (float*)d_out;

  const int N = in_sizes[0] / 3;             // V is [N,3]
  const int waves_per_block = 256 / 32;      // 8 vertices per block
  const int blocks = (N + waves_per_block - 1) / waves_per_block;

  if (ws_size >= (size_t)N * sizeof(float)) {
    float* pv = (float*)d_ws;
    arap_vertex_kernel<<<blocks, 256, 0, stream>>>(V, Vd, nbrs, wgts, pv, out,
                                                   N, 0);
    arap_reduce_kernel<<<1, 256, 0, stream>>>(pv, out, N);
  } else {
    hipMemsetAsync(out, 0, sizeof(float), stream);
    arap_vertex_kernel<<<blocks, 256, 0, stream>>>(V, Vd, nbrs, wgts, nullptr,
                                                   out, N, 1);
  }
}